// LocallyConnected2d_66142496358597
// MI455X (gfx1250) — compile-verified
//
#include <hip/hip_runtime.h>

typedef __attribute__((ext_vector_type(2))) float v2f;
typedef __attribute__((ext_vector_type(4))) float v4f;
typedef __attribute__((ext_vector_type(8))) float v8f;

#define B_  32
#define IC  32
#define OC  32
#define HH  64
#define WW  64
#define KD  288        // IC * 3 * 3 flattened contraction dim
#define LDP 292        // padded LDS row stride (floats): breaks 64-bank conflicts, 16B-preserving

__global__ __launch_bounds__(128)
void lc2d_wmma_f32_kernel(const float* __restrict__ x,
                          const float* __restrict__ wgt,
                          const float* __restrict__ bias,
                          float* __restrict__ out) {
    __shared__ float sA[B_ * LDP];  // patches:  [b][c*9 + i*3 + j]
    __shared__ float sB[OC * LDP];  // weights:  [o][c*9 + i*3 + j]

    const int w  = blockIdx.x;
    const int h  = blockIdx.y;
    const int t  = threadIdx.x;
    const int hw = h * WW + w;

    // ---- stage this location's weight slice: 32x288, contiguous, 128-bit + non-temporal ----
    // 288 % 4 == 0 and 292 % 4 == 0 -> float4 never crosses a row, LDS addr stays 16B aligned.
    {
        const v4f* wp4 = (const v4f*)(wgt + (size_t)hw * (OC * KD));
        for (int e4 = t; e4 < (OC * KD) / 4; e4 += 128) {
            int e   = e4 * 4;
            int row = e / KD;
            int col = e - row * KD;
            v4f v = __builtin_nontemporal_load(wp4 + e4);  // TH=NT: streamed once, keep L2 for x
            *(v4f*)&sB[row * LDP + col] = v;
        }
    }

    // ---- stage input patches, row-granular: (b,c,i) -> 3 contiguous x elements (j=0..2) ----
    for (int r = t; r < B_ * IC * 3; r += 128) {
        int b   = r / 96;
        int rem = r - b * 96;
        int c   = rem / 3;
        int i   = rem - c * 3;
        int y   = h - 1 + i;
        const bool yok = (unsigned)y < (unsigned)HH;
        const float* xrow = x + ((((size_t)b * IC + c) * HH + y) * WW + (w - 1));
        float* dst = &sA[b * LDP + c * 9 + i * 3];
        dst[0] = (yok && w - 1 >= 0) ? xrow[0] : 0.0f;
        dst[1] = yok                 ? xrow[1] : 0.0f;
        dst[2] = (yok && w + 1 < WW) ? xrow[2] : 0.0f;
    }
    __syncthreads();

    // ---- 32x32x288 GEMM via chained V_WMMA_F32_16X16X4_F32 ----
    const int wv    = t >> 5;          // wave 0..3 -> one 16x16 tile each
    const int lane  = t & 31;
    const int m0    = (wv >> 1) * 16;  // batch tile base
    const int n0    = (wv & 1) * 16;   // out-channel tile base
    const int lr    = lane & 15;
    const int khalf = lane >> 4;       // 0: K-pair {0,1}; 1: K-pair {2,3}

    const float* aRow = &sA[(m0 + lr) * LDP + 2 * khalf];
    const float* bRow = &sB[(n0 + lr) * LDP + 2 * khalf];

    v8f acc = {};
    #pragma unroll 8
    for (int kt = 0; kt < KD / 4; ++kt) {
        v2f a = *(const v2f*)(aRow + kt * 4);
        v2f b = *(const v2f*)(bRow + kt * 4);
        // (neg_a, A, neg_b, B, c_mod, C, reuse_a, reuse_b)
        acc = __builtin_amdgcn_wmma_f32_16x16x4_f32(
            false, a, false, b, (short)0, acc, false, false);
    }

    // ---- epilogue: C/D layout VGPR r -> lanes 0-15: M=r, lanes 16-31: M=r+8 ----
    const int N  = lane & 15;
    const int o  = n0 + N;
    const float bv = bias[(size_t)o * (HH * WW) + hw];
    #pragma unroll
    for (int r = 0; r < 8; ++r) {
        int M = r + khalf * 8;
        int b = m0 + M;
        // write-once output: non-temporal store, don't pollute L2
        __builtin_nontemporal_store(acc[r] + bv,
                                    out + (((size_t)b * OC + o) * (HH * WW)) + hw);
    }
}

extern "C" void kernel_launch(void* const* d_in, const int* in_sizes, int n_in,
                              void* d_out, int out_size, void* d_ws, size_t ws_size,
                              hipStream_t stream) {
    const float* x    = (const float*)d_in[0];
    const float* wgt  = (const float*)d_in[1];
    const float* bias = (const float*)d_in[2];
    float* out        = (float*)d_out;
    (void)in_sizes; (void)n_in; (void)out_size; (void)d_ws; (void)ws_size;

    dim3 grid(WW, HH);   // one block per spatial location
    lc2d_wmma_f32_kernel<<<grid, 128, 0, stream>>>(x, wgt, bias, out);
}